// GraphSAGE_51161650430634
// MI455X (gfx1250) — compile-verified
//
#include <hip/hip_runtime.h>

typedef __attribute__((ext_vector_type(2))) float v2f;
typedef __attribute__((ext_vector_type(8))) float v8f;

#define HF 128  // feature width (IN == H == 128)

// ---------------------------------------------------------------------------
// Zero fill (graph-capture-safe replacement for memset)
// ---------------------------------------------------------------------------
__global__ void fill_zero_kernel(float* __restrict__ p, int n) {
  int i = blockIdx.x * blockDim.x + threadIdx.x;
  if (i < n) p[i] = 0.0f;
}

// ---------------------------------------------------------------------------
// Fold SAGE bias + BatchNorm(eval) into per-column scale/bias:
//   y_bn = (y + bl - m) * g/sqrt(v+eps) + b  =  y*scale + bias
// ---------------------------------------------------------------------------
__global__ void fold_bn_kernel(const float* __restrict__ bl, const float* __restrict__ g,
                               const float* __restrict__ b,  const float* __restrict__ m,
                               const float* __restrict__ v,  float* __restrict__ scale,
                               float* __restrict__ bias) {
  int j = threadIdx.x;  // 128 threads
  float s = g[j] * rsqrtf(v[j] + 1e-5f);
  scale[j] = s;
  bias[j]  = (bl[j] - m[j]) * s + b[j];
}

// ---------------------------------------------------------------------------
// Pack combined weight  B[k][j] = (k<128 ? Wl[j][k] : Wr[j][k-128])  (K=256, N=128)
// into V_WMMA_F32_16X16X4_F32 B-fragment order:
//   Bp[((ntile*64 + kstep)*32 + lane)*2 + t] = B[kstep*4 + ((lane>=16)?2:0) + t][ntile*16 + lane%16]
// ---------------------------------------------------------------------------
__global__ void pack_b_kernel(const float* __restrict__ Wl, const float* __restrict__ Wr,
                              float* __restrict__ Bp) {
  int t = blockIdx.x * blockDim.x + threadIdx.x;  // 16384 total
  if (t >= 16384) return;
  int lane  = t & 31;
  int kstep = (t >> 5) & 63;
  int ntile = t >> 11;
  int col = ntile * 16 + (lane & 15);
  int k0  = kstep * 4 + ((lane >> 4) << 1);   // k0 is even -> never straddles the 128 split
  int k1  = k0 + 1;
  float vx = (k0 < HF) ? Wl[col * HF + k0] : Wr[col * HF + (k0 - HF)];
  float vy = (k1 < HF) ? Wl[col * HF + k1] : Wr[col * HF + (k1 - HF)];
  Bp[t * 2 + 0] = vx;
  Bp[t * 2 + 1] = vy;
}

// ---------------------------------------------------------------------------
// Edge scatter: one wave (32 lanes) per edge; each lane adds 4 features.
// ---------------------------------------------------------------------------
__global__ void sage_aggregate_kernel(const float* __restrict__ X,
                                      const int* __restrict__ src,
                                      const int* __restrict__ dst,
                                      float* __restrict__ agg,
                                      float* __restrict__ cnt, int E) {
  int e    = (blockIdx.x * blockDim.x + threadIdx.x) >> 5;
  int lane = threadIdx.x & 31;
  if (e >= E) return;
  int s = src[e];
  int d = dst[e];
  const float4* xr = (const float4*)(X + (size_t)s * HF);
  float4 vv = xr[lane];
  float* ar = agg + (size_t)d * HF + lane * 4;
  atomicAdd(ar + 0, vv.x);
  atomicAdd(ar + 1, vv.y);
  atomicAdd(ar + 2, vv.z);
  atomicAdd(ar + 3, vv.w);
  if (lane == 0) atomicAdd(cnt + d, 1.0f);
}

// ---------------------------------------------------------------------------
// Fused SAGE layer GEMM + BN + ReLU using V_WMMA_F32_16X16X4_F32.
//   Out[n,:] = relu( ([agg[n]/max(cnt,1) | X[n]] @ Bc) * scale + bias )
// One wave per 16-row M-tile, full 128 output columns (8 N-tiles).
// K = 256 -> 64 wmma steps of K=4, x8 N-tiles = 512 wmma per wave.
// ---------------------------------------------------------------------------
__global__ __launch_bounds__(256)
void sage_gemm_kernel(const float* __restrict__ agg, const float* __restrict__ X,
                      const float* __restrict__ cnt, const float* __restrict__ Bp,
                      const float* __restrict__ scale, const float* __restrict__ bias,
                      float* __restrict__ Out, int mtiles) {
  int lane  = threadIdx.x & 31;
  int mtile = blockIdx.x * 8 + (threadIdx.x >> 5);
  if (mtile >= mtiles) return;  // wave-uniform: EXEC stays all-ones for WMMA

  int row  = mtile * 16 + (lane & 15);
  int koff = (lane >> 4) << 1;  // lanes 0-15 hold K=0,1 ; lanes 16-31 hold K=2,3
  float inv = 1.0f / fmaxf(cnt[row], 1.0f);

  const float* aB = agg + (size_t)row * HF + koff;
  const float* xB = X   + (size_t)row * HF + koff;
  const v2f*   bp = (const v2f*)Bp;

  v8f acc[8] = {};

  // K = 0..127 : normalized neighbor aggregate (lin_l input)
  #pragma unroll 4
  for (int ks = 0; ks < 32; ++ks) {
    v2f a = *(const v2f*)(aB + ks * 4);
    a *= inv;
    #pragma unroll
    for (int nt = 0; nt < 8; ++nt) {
      v2f b = bp[(nt * 64 + ks) * 32 + lane];
      acc[nt] = __builtin_amdgcn_wmma_f32_16x16x4_f32(
          false, a, false, b, (short)0, acc[nt], false, false);
    }
  }
  // K = 128..255 : root features (lin_r input)
  #pragma unroll 4
  for (int ks = 0; ks < 32; ++ks) {
    v2f a = *(const v2f*)(xB + ks * 4);
    #pragma unroll
    for (int nt = 0; nt < 8; ++nt) {
      v2f b = bp[(nt * 64 + (32 + ks)) * 32 + lane];
      acc[nt] = __builtin_amdgcn_wmma_f32_16x16x4_f32(
          false, a, false, b, (short)0, acc[nt], false, false);
    }
  }

  // Epilogue: BN (folded) + ReLU, then store.
  // C layout: VGPR i -> M = i (lanes 0-15) / M = 8+i (lanes 16-31), N = lane%16.
  int rbase = mtile * 16 + ((lane >> 4) << 3);
  int cbase = lane & 15;
  #pragma unroll
  for (int nt = 0; nt < 8; ++nt) {
    int col = nt * 16 + cbase;
    float s  = scale[col];
    float bb = bias[col];
    #pragma unroll
    for (int i = 0; i < 8; ++i) {
      float y = fmaxf(acc[nt][i] * s + bb, 0.0f);
      Out[(size_t)(rbase + i) * HF + col] = y;
    }
  }
}

// ---------------------------------------------------------------------------
// Classification head: out[n,:2] = h[n] @ Wlin^T + blin   (OUT=2, trivial VALU)
// ---------------------------------------------------------------------------
__global__ void head_kernel(const float* __restrict__ hfeat, const float* __restrict__ Wlin,
                            const float* __restrict__ blin, float* __restrict__ out, int n) {
  int i = blockIdx.x * blockDim.x + threadIdx.x;
  if (i >= n) return;
  const float4* hr = (const float4*)(hfeat + (size_t)i * HF);
  const float4* w0 = (const float4*)(Wlin);
  const float4* w1 = (const float4*)(Wlin + HF);
  float a0 = 0.0f, a1 = 0.0f;
  #pragma unroll
  for (int q = 0; q < HF / 4; ++q) {
    float4 hv = hr[q], u = w0[q], w = w1[q];
    a0 += hv.x * u.x + hv.y * u.y + hv.z * u.z + hv.w * u.w;
    a1 += hv.x * w.x + hv.y * w.y + hv.z * w.z + hv.w * w.w;
  }
  out[(size_t)i * 2 + 0] = a0 + blin[0];
  out[(size_t)i * 2 + 1] = a1 + blin[1];
}

// ---------------------------------------------------------------------------
extern "C" void kernel_launch(void* const* d_in, const int* in_sizes, int n_in,
                              void* d_out, int out_size, void* d_ws, size_t ws_size,
                              hipStream_t stream) {
  const float* x    = (const float*)d_in[0];
  const int*   ei   = (const int*)d_in[1];   // [2, E]
  const float* Wl1  = (const float*)d_in[2];
  const float* bl1  = (const float*)d_in[3];
  const float* Wr1  = (const float*)d_in[4];
  const float* Wl2  = (const float*)d_in[5];
  const float* bl2  = (const float*)d_in[6];
  const float* Wr2  = (const float*)d_in[7];
  const float* g1   = (const float*)d_in[8];
  const float* b1   = (const float*)d_in[9];
  const float* m1   = (const float*)d_in[10];
  const float* v1   = (const float*)d_in[11];
  const float* g2   = (const float*)d_in[12];
  const float* b2   = (const float*)d_in[13];
  const float* m2   = (const float*)d_in[14];
  const float* v2   = (const float*)d_in[15];
  const float* Wlin = (const float*)d_in[16];
  const float* blin = (const float*)d_in[17];
  float* out = (float*)d_out;

  const int Nn = in_sizes[0] / HF;   // 50000
  const int E  = in_sizes[1] / 2;    // 800000
  const int* src = ei;
  const int* dst = ei + E;

  // Workspace carve (256B aligned)
  char* base = (char*)d_ws;
  size_t off = 0;
  auto carve = [&](size_t bytes) -> float* {
    float* p = (float*)(base + off);
    off = (off + bytes + 255) & ~(size_t)255;
    return p;
  };
  float* Bp1  = carve(32768 * sizeof(float));
  float* Bp2  = carve(32768 * sizeof(float));
  float* sc1  = carve(HF * sizeof(float));
  float* bi1  = carve(HF * sizeof(float));
  float* sc2  = carve(HF * sizeof(float));
  float* bi2  = carve(HF * sizeof(float));
  float* cnt  = carve((size_t)Nn * sizeof(float));
  float* agg  = carve((size_t)Nn * HF * sizeof(float));
  float* hbuf = carve((size_t)Nn * HF * sizeof(float));

  const int mtiles = Nn / 16;                       // 3125 (exact)
  const int gemmBlocks = (mtiles + 7) / 8;
  const int aggThreads = E * 32;

  // Weight / BN preprocessing (tiny)
  pack_b_kernel<<<64, 256, 0, stream>>>(Wl1, Wr1, Bp1);
  pack_b_kernel<<<64, 256, 0, stream>>>(Wl2, Wr2, Bp2);
  fold_bn_kernel<<<1, HF, 0, stream>>>(bl1, g1, b1, m1, v1, sc1, bi1);
  fold_bn_kernel<<<1, HF, 0, stream>>>(bl2, g2, b2, m2, v2, sc2, bi2);

  // Layer 1
  fill_zero_kernel<<<(Nn * HF + 255) / 256, 256, 0, stream>>>(agg, Nn * HF);
  fill_zero_kernel<<<(Nn + 255) / 256, 256, 0, stream>>>(cnt, Nn);
  sage_aggregate_kernel<<<(aggThreads + 255) / 256, 256, 0, stream>>>(x, src, dst, agg, cnt, E);
  sage_gemm_kernel<<<gemmBlocks, 256, 0, stream>>>(agg, x, cnt, Bp1, sc1, bi1, hbuf, mtiles);

  // Layer 2 (GEMM writes hbuf in place: each wave reads only its own rows first)
  fill_zero_kernel<<<(Nn * HF + 255) / 256, 256, 0, stream>>>(agg, Nn * HF);
  fill_zero_kernel<<<(Nn + 255) / 256, 256, 0, stream>>>(cnt, Nn);
  sage_aggregate_kernel<<<(aggThreads + 255) / 256, 256, 0, stream>>>(hbuf, src, dst, agg, cnt, E);
  sage_gemm_kernel<<<gemmBlocks, 256, 0, stream>>>(agg, hbuf, cnt, Bp2, sc2, bi2, hbuf, mtiles);

  // Head
  head_kernel<<<(Nn + 255) / 256, 256, 0, stream>>>(hbuf, Wlin, blin, out, Nn);
}